// SkipGramNegativeSamplingModel_20023137534890
// MI455X (gfx1250) — compile-verified
//
#include <hip/hip_runtime.h>
#include <math.h>

// CDNA5 (gfx1250) wave32 WMMA implementation of SkipGram-NS loss.
// z[p] = dot(emb[src[p]], cls_w[ctx[p]]) + cls_b[ctx[p]]
// loss = mean(softplus(z) - z*y)
//
// Dot products are computed 16-at-a-time on the matrix pipe with
// V_WMMA_F32_16X16X4_F32 (f32 in / f32 accum -> bit-comparable precision to
// the f32 reference); the 16 needed dots are the diagonal of the 16x16 tile.

typedef __attribute__((ext_vector_type(2))) float v2f;
typedef __attribute__((ext_vector_type(8))) float v8f;

constexpr int Dim    = 128;
constexpr int NPAIR  = 512 * 5;        // B * L = 2560
constexpr int TILES  = NPAIR / 16;     // 160

__global__ __launch_bounds__(32)
void sgns_wmma_tile_kernel(const float* __restrict__ emb,
                           const float* __restrict__ clsw,
                           const float* __restrict__ clsb,
                           const int*   __restrict__ src,
                           const int*   __restrict__ ctx,
                           const int*   __restrict__ lab,
                           float*       __restrict__ partial) {
    const int tile = blockIdx.x;
    const int lane = threadIdx.x;          // 0..31 (wave32)
    const int r    = lane & 15;            // matrix row this lane serves
    const int hi   = lane >> 4;            // 0: K={0,1}, 1: K={2,3} per step
    const int p    = tile * 16 + r;        // pair index

    const int si = src[p];
    const int ci = ctx[p];
    const float* __restrict__ xrow = emb  + (size_t)si * Dim;
    const float* __restrict__ wrow = clsw + (size_t)ci * Dim;
    const int colbase = 2 * hi;

    // Accumulate C += A(16x4) * B(4x16) over K = 0..127 in steps of 4.
    v8f acc = {};
    #pragma unroll
    for (int k = 0; k < Dim; k += 4) {
        v2f a, b;
        a.x = xrow[k + colbase + 0];
        a.y = xrow[k + colbase + 1];
        b.x = wrow[k + colbase + 0];
        b.y = wrow[k + colbase + 1];
        // 8-arg form: (neg_a, A, neg_b, B, c_mod, C, reuse_a, reuse_b)
        acc = __builtin_amdgcn_wmma_f32_16x16x4_f32(
            false, a, false, b, (short)0, acc, false, false);
    }

    // Diagonal element (m,m) of the 16x16 f32 C/D layout:
    //   m = 0..7  -> lane m      , VGPR m
    //   m = 8..15 -> lane m + 16 , VGPR m - 8
    // so each diag value sits in a distinct lane; m == lane & 15 there.
    const bool active = (lane < 8) || (lane >= 24);
    const int  idx    = ((lane < 8) ? lane : (lane - 24)) & 7;
    float e = acc[0];
    #pragma unroll
    for (int j = 1; j < 8; ++j) e = (idx == j) ? acc[j] : e;

    const float z = e + clsb[ci];
    const float y = (float)lab[p];
    // numerically stable softplus(z) - z*y
    float contrib = fmaxf(z, 0.0f) + log1pf(expf(-fabsf(z))) - z * y;
    contrib = active ? contrib : 0.0f;

    // wave32 butterfly reduction (16 live values -> lane 0)
    #pragma unroll
    for (int off = 16; off > 0; off >>= 1)
        contrib += __shfl_xor(contrib, off, 32);

    if (lane == 0) partial[tile] = contrib;
}

// Fixed-order final reduction: deterministic across replays (no float atomics).
__global__ void sgns_reduce_kernel(const float* __restrict__ partial,
                                   float* __restrict__ out) {
    if (threadIdx.x == 0 && blockIdx.x == 0) {
        float s = 0.0f;
        for (int i = 0; i < TILES; ++i) s += partial[i];
        out[0] = s / (float)NPAIR;
    }
}

extern "C" void kernel_launch(void* const* d_in, const int* in_sizes, int n_in,
                              void* d_out, int out_size, void* d_ws, size_t ws_size,
                              hipStream_t stream) {
    const float* emb  = (const float*)d_in[0];   // [V, D] f32
    const float* clsw = (const float*)d_in[1];   // [V, D] f32
    const float* clsb = (const float*)d_in[2];   // [V]    f32
    const int*   src  = (const int*)d_in[3];     // [B, L] i32
    const int*   ctx  = (const int*)d_in[4];     // [B, L] i32
    const int*   lab  = (const int*)d_in[5];     // [B, L] i32
    float* partial = (float*)d_ws;               // 160 floats of scratch
    float* out     = (float*)d_out;

    sgns_wmma_tile_kernel<<<TILES, 32, 0, stream>>>(emb, clsw, clsb, src, ctx, lab, partial);
    sgns_reduce_kernel<<<1, 32, 0, stream>>>(partial, out);
}